// AdvancedVibLayer_2276332667555
// MI455X (gfx1250) — compile-verified
//
#include <hip/hip_runtime.h>
#include <math.h>

typedef __attribute__((ext_vector_type(2))) float v2f;
typedef __attribute__((ext_vector_type(8))) float v8f;

#define NSTEPS 6
#define DTC 0.2f
#define CC 64
#define HH 256
#define WW 256
#define TILE 64     // output tile
#define HALO 8      // computed halo (>= NSTEPS)
#define LDS_W 84    // cols -2..81 of 80-wide compute region
#define LDS_H 82    // rows -1..80

__global__ __launch_bounds__(160) void vib_wmma_kernel(
    const float* __restrict__ force,
    const float* __restrict__ cw,
    const float* __restrict__ omega,
    const float* __restrict__ zeta,
    float* __restrict__ out)
{
  __shared__ __align__(16) float xs[LDS_H * LDS_W];

  const int g    = blockIdx.x;
  const int tile = g & 15;
  const int c    = (g >> 4) & (CC - 1);
  const int bb   = g >> 10;
  const int tx   = tile & 3, ty = tile >> 2;

  const int tid  = threadIdx.x;
  const int wave = tid >> 5;
  const int lane = tid & 31;
  const int lm   = lane & 15;        // M (A rows) or N (B/C cols) within 16
  const int hi   = (lane >> 4) & 1;  // lanes 16-31 carry K+2 (A/B) or M+8 (C/D)

  // zero LDS x (permanent zero ring = conv zero padding + halo margin)
  for (int i = tid; i < LDS_H * LDS_W; i += 160) xs[i] = 0.0f;

  // per-channel parameters (uniform across workgroup)
  const float om = omega[c];
  const float zt = zeta[c];
  const float wn = log1pf(__expf(om));            // softplus
  const float zz = 1.0f / (1.0f + __expf(-zt));   // sigmoid
  const float w2 = wn * wn;
  const float a1 = -2.0f * zz * wn;
  const float b1 = -w2;

  float w3[3][3];
  #pragma unroll
  for (int r = 0; r < 3; ++r)
    #pragma unroll
    for (int d = 0; d < 3; ++d)
      w3[r][d] = cw[c * 9 + r * 3 + d];

  // Banded B fragments (constant over steps). K=20 (5 chunks of 4) covers
  // x columns colbase-2 .. colbase+17, so the horizontal halo is inside the
  // matmul. B[k][n] = w3[r][k-2-n+1] when |k-2-n|<=1, else 0.
  // Lane layout per chunk: .x -> K = 4*ch + 2*hi, .y -> K+1, n = lm.
  v2f Bf[3][5];
  #pragma unroll
  for (int r = 0; r < 3; ++r) {
    #pragma unroll
    for (int ch = 0; ch < 5; ++ch) {
      v2f bv;
      #pragma unroll
      for (int h = 0; h < 2; ++h) {
        int k = 4 * ch + 2 * hi + h;
        int d = k - 2 - lm;
        bv[h] = (d >= -1 && d <= 1) ? w3[r][d + 1] : 0.0f;
      }
      Bf[r][ch] = bv;
    }
  }

  // Per-block state in C/D layout: element e -> row 16*wave + e + 8*hi,
  // col 16*b + lm of the 80x80 compute region.
  v8f xr[5], vr[5], fr[5];
  unsigned mb[5];
  const int gy0 = ty * TILE - HALO;
  const int gx0 = tx * TILE - HALO;
  const long base = ((long)bb * CC + c) * (long)(HH * WW);

  #pragma unroll
  for (int b = 0; b < 5; ++b) {
    unsigned m = 0;
    #pragma unroll
    for (int e = 0; e < 8; ++e) {
      int lr = 16 * wave + e + 8 * hi;
      int lc = 16 * b + lm;
      int gy = gy0 + lr, gx = gx0 + lc;
      bool ok = (gy >= 0) && (gy < HH) && (gx >= 0) && (gx < WW);
      float f = 0.0f;
      if (ok) { f = force[base + (long)gy * WW + gx]; m |= (1u << e); }
      fr[b][e] = f;
      xr[b][e] = 0.0f;
      vr[b][e] = 0.0f;
    }
    mb[b] = m;
  }

  __syncthreads();

  for (int s = 0; s < NSTEPS; ++s) {
    #pragma unroll
    for (int b = 0; b < 5; ++b) {
      v8f D = {};
      #pragma unroll
      for (int r = 0; r < 3; ++r) {
        int arow = 16 * wave + lm + r;  // row-shifted A fragment (Y_r)
        #pragma unroll
        for (int ch = 0; ch < 5; ++ch) {
          int acol = 16 * b + 4 * ch + 2 * hi;  // even -> 8B aligned
          v2f A = *(const v2f*)&xs[arow * LDS_W + acol];
          D = __builtin_amdgcn_wmma_f32_16x16x4_f32(
                false, A, false, Bf[r][ch], (short)0, D, false, false);
      }
      }
      // leapfrog update, masked so out-of-image pixels stay exactly zero
      #pragma unroll
      for (int e = 0; e < 8; ++e) {
        float x = xr[b][e], v = vr[b][e];
        float dv = (a1 * v + b1 * x + fr[b][e] + D[e]) * DTC;
        v += dv;
        x += v * DTC;
        bool ok = (mb[b] >> e) & 1u;
        vr[b][e] = ok ? v : 0.0f;
        xr[b][e] = ok ? x : 0.0f;
      }
    }
    __syncthreads();  // all halo reads of this step done
    #pragma unroll
    for (int b = 0; b < 5; ++b) {
      #pragma unroll
      for (int e = 0; e < 8; ++e) {
        int arow = 16 * wave + e + 8 * hi + 1;  // +1: LDS row origin at -1
        int acol = 16 * b + lm + 2;             // +2: LDS col origin at -2
        xs[arow * LDS_W + acol] = xr[b][e];
      }
    }
    __syncthreads();  // new x visible before next step's conv
  }

  // energy of the central 64x64 core (each output written exactly once)
  #pragma unroll
  for (int b = 0; b < 5; ++b) {
    #pragma unroll
    for (int e = 0; e < 8; ++e) {
      int lr = 16 * wave + e + 8 * hi;
      int lc = 16 * b + lm;
      if (lr >= HALO && lr < HALO + TILE && lc >= HALO && lc < HALO + TILE) {
        int gy = gy0 + lr, gx = gx0 + lc;
        float x = xr[b][e], v = vr[b][e];
        out[base + (long)gy * WW + gx] = 0.5f * v * v + 0.5f * w2 * x * x;
      }
    }
  }
}

extern "C" void kernel_launch(void* const* d_in, const int* in_sizes, int n_in,
                              void* d_out, int out_size, void* d_ws, size_t ws_size,
                              hipStream_t stream) {
  const float* force = (const float*)d_in[0];
  const float* cw    = (const float*)d_in[1];
  const float* omg   = (const float*)d_in[2];
  const float* zet   = (const float*)d_in[3];
  float* out = (float*)d_out;

  int nB = in_sizes[0] / (CC * HH * WW);  // batch from flat size
  dim3 grid(nB * CC * 16);
  dim3 block(160);
  hipLaunchKernelGGL(vib_wmma_kernel, grid, block, 0, stream,
                     force, cw, omg, zet, out);
}